// ResidualCodebookCollection_77824807403890
// MI455X (gfx1250) — compile-verified
//
#include <hip/hip_runtime.h>

typedef float v2f __attribute__((ext_vector_type(2)));
typedef float v8f __attribute__((ext_vector_type(8)));

// Problem constants (match reference setup_inputs)
constexpr int Bsz = 32;     // batch
constexpr int Dd  = 64;     // feature dim
constexpr int Ll  = 4096;   // sequence length
constexpr int NCB = 4;      // codebooks
constexpr int Kc  = 512;    // codes per codebook

constexpr int STR = 68;     // padded LDS row stride (floats): bank-conflict-free, 16B-aligned rows
constexpr int TOK_PER_BLOCK = 128;                 // 8 waves * 16 tokens
constexpr int BLK_PER_B     = Ll / TOK_PER_BLOCK;  // 32

__global__ __launch_bounds__(256)
void rvq_wmma_kernel(const float* __restrict__ x,
                     const float* __restrict__ cb,
                     float* __restrict__ out)
{
  // Whole codebook staged in LDS once per codebook: 512*68*4 = 136 KB (< 320 KB/WGP).
  __shared__ __align__(16) float s_cb[Kc][STR];
  __shared__ float s_enorm[Kc];      // per-code squared norms
  __shared__ int   s_idx[8][16];     // winning code per token, per wave

  const int tid  = threadIdx.x;
  const int w    = tid >> 5;
  const int lane = tid & 31;
  const int m    = lane & 15;        // A: token row M ; B/D: column N
  const int hi   = lane >> 4;        // half-wave select
  const int koff = hi * 2;           // f32 A/B fragment K offset per half

  const int b   = blockIdx.x >> 5;                 // 32 blocks per batch row
  const int l0b = (blockIdx.x & (BLK_PER_B - 1)) * TOK_PER_BLOCK;
  const int lw  = l0b + w * 16;                    // wave's token base
  const int l   = lw + m;                          // this lane's token (column in L)

  // Residual tile in WMMA-A fragment layout, plus cumulative z_q accumulator.
  v2f afrag[16];
  v2f zacc[16];
  const float* xb = x + (size_t)b * Dd * Ll;
#pragma unroll
  for (int kk = 0; kk < 16; ++kk) {
    const int d = 4 * kk + koff;
    afrag[kk].x = xb[(size_t)d       * Ll + l];
    afrag[kk].y = xb[(size_t)(d + 1) * Ll + l];
    zacc[kk].x = 0.f;
    zacc[kk].y = 0.f;
  }

  for (int c = 0; c < NCB; ++c) {
    const float* cbc = cb + (size_t)c * Kc * Dd;

    __syncthreads();   // protect s_cb/s_idx from previous iteration's readers
    // Stage entire codebook c into LDS (coalesced b128 global loads, b128 LDS stores).
    for (int i = tid; i < Kc * (Dd / 4); i += 256) {
      const int row = i >> 4;
      const int f4  = (i & 15) << 2;
      const float4 v = *(const float4*)(cbc + (size_t)row * Dd + f4);
      *(float4*)&s_cb[row][f4] = v;
    }
    __syncthreads();
    // Per-code squared norms from the LDS copy.
    for (int code = tid; code < Kc; code += 256) {
      float s = 0.f;
#pragma unroll 8
      for (int dd = 0; dd < Dd; ++dd) s += s_cb[code][dd] * s_cb[code][dd];
      s_enorm[code] = s;
    }
    __syncthreads();

    float best[8];
    int   bestn[8];
#pragma unroll
    for (int r = 0; r < 8; ++r) { best[r] = 3.4e38f; bestn[r] = 0; }

    // Barrier-free sweep over all 32 N-tiles; B fragments straight from LDS.
    for (int nt = 0; nt < Kc / 16; ++nt) {
      const int n0 = nt * 16;
      const float* bsrc = &s_cb[n0 + m][0];   // this lane's code row

      v8f acc = {};
#pragma unroll
      for (int kk = 0; kk < 16; ++kk) {
        v2f bf;                               // B fragment: lane = code column
        bf.x = bsrc[4 * kk + koff];
        bf.y = bsrc[4 * kk + koff + 1];
        acc = __builtin_amdgcn_wmma_f32_16x16x4_f32(
            /*neg_a=*/false, afrag[kk], /*neg_b=*/false, bf,
            /*c_mod=*/(short)0, acc, /*reuse_a=*/false, /*reuse_b=*/false);
      }

      const float en    = s_enorm[n0 + m];    // column-constant across rows
      const int   ncode = n0 + m;
#pragma unroll
      for (int r = 0; r < 8; ++r) {
        const float score = en - 2.f * acc[r];  // argmin of d2 minus const ||x||^2
        if (score < best[r]) { best[r] = score; bestn[r] = ncode; }
      }
    }

    // Cross-lane argmin within each 16-lane half (columns of the D tile).
#pragma unroll
    for (int r = 0; r < 8; ++r) {
      float v = best[r];
      int   i = bestn[r];
#pragma unroll
      for (int off = 8; off >= 1; off >>= 1) {
        const float v2 = __shfl_xor(v, off, 32);
        const int   i2 = __shfl_xor(i, off, 32);
        if (v2 < v || (v2 == v && i2 < i)) { v = v2; i = i2; }
      }
      const int mr = r + 8 * hi;              // token row for this half
      if (m == 0) {
        s_idx[w][mr] = i;
        // indices output [B, L, NCB], stored as float after the z_q block
        out[(size_t)Bsz * NCB * Dd * Ll +
            ((size_t)b * Ll + (lw + mr)) * NCB + c] = (float)i;
      }
    }

    // Residual update + cumulative z_q output; winning code row gathered from LDS.
    const int myidx = s_idx[w][m];            // same-wave LDS write->read (in-order DS)
    const float* erow = &s_cb[myidx][0];
    float* zo = out + (((size_t)b * NCB + c) * Dd) * (size_t)Ll;
#pragma unroll
    for (int kk = 0; kk < 16; ++kk) {
      const int d = 4 * kk + koff;
      const float ex = erow[d];
      const float ey = erow[d + 1];
      afrag[kk].x -= ex;  afrag[kk].y -= ey;
      zacc[kk].x  += ex;  zacc[kk].y  += ey;
      zo[(size_t)d       * Ll + l] = zacc[kk].x;
      zo[(size_t)(d + 1) * Ll + l] = zacc[kk].y;
    }
  }
}

extern "C" void kernel_launch(void* const* d_in, const int* in_sizes, int n_in,
                              void* d_out, int out_size, void* d_ws, size_t ws_size,
                              hipStream_t stream) {
  (void)in_sizes; (void)n_in; (void)out_size; (void)d_ws; (void)ws_size;
  const float* x  = (const float*)d_in[0];   // [32, 64, 4096]
  const float* cb = (const float*)d_in[1];   // [4, 512, 64]
  float* out = (float*)d_out;                // [32,4,64,4096] f32 ++ [32,4096,4] idx-as-float

  const dim3 grid(Bsz * BLK_PER_B);          // 1024 blocks
  const dim3 block(256);                     // 8 wave32
  rvq_wmma_kernel<<<grid, block, 0, stream>>>(x, cb, out);
}